// GIN_60404420051054
// MI455X (gfx1250) — compile-verified
//
#include <hip/hip_runtime.h>

typedef __attribute__((ext_vector_type(2))) float v2f;
typedef __attribute__((ext_vector_type(8))) float v8f;

#define NNODES 100000
#define NEDGES 1600000
#define NGRAPH 64
#define MTILES (NNODES / 16)   // 6250, exact
#define BN_EPS 1e-5f
#define SW_STRIDE 66           // padded LDS stride (8B-aligned, conflict-friendly)

__device__ __forceinline__ void gatomicAdd(float* p, float v) {
    // lowers to global_atomic_add_f32 on gfx1250 (no CAS loop)
    unsafeAtomicAdd(p, v);
}

// ---------------- utility kernels ----------------

__global__ void zero_kernel(float* __restrict__ stats, float* __restrict__ pool,
                            float* __restrict__ counts) {
    int t = blockIdx.x * 256 + threadIdx.x;
    if (t < 128)  stats[t]  = 0.f;
    if (t < 4096) pool[t]   = 0.f;
    if (t < 64)   counts[t] = 0.f;
}

__global__ void copy_kernel(const float4* __restrict__ src, float4* __restrict__ dst, int n4) {
    int t = blockIdx.x * 256 + threadIdx.x;
    if (t < n4) dst[t] = src[t];
}

// z[dst] += x[src] over all edges; 2 features per lane (float2)
__global__ void scatter_kernel(const float* __restrict__ x,
                               const long long* __restrict__ ei,
                               float* __restrict__ z) {
    int t  = blockIdx.x * 256 + threadIdx.x;
    int e  = t >> 5;
    int fp = t & 31;
    if (e >= NEDGES) return;
    long long s = ei[e];
    long long d = ei[NEDGES + e];
    float2 v = ((const float2*)(x + s * 64))[fp];
    float* zp = z + d * 64 + fp * 2;
    gatomicAdd(zp,     v.x);
    gatomicAdd(zp + 1, v.y);
}

// ---------------- fused WMMA GEMM ----------------
// Out = act(Zin) @ W + bias, where act = identity or relu(x*scale+shift).
// Optionally accumulates per-feature sum / sum-of-squares into stats[0:64]/[64:128].
__global__ __launch_bounds__(256) void gemm_kernel(
    const float* __restrict__ Zin, const float* __restrict__ W,
    const float* __restrict__ bias,
    const float* __restrict__ scale, const float* __restrict__ shift,
    float* __restrict__ Out, float* __restrict__ stats,
    int doBN, int doStats)
{
    __shared__ float sW[64 * SW_STRIDE];   // W transposed: sW[n*66 + k] = W[k][n]
    __shared__ float sStats[128];

    int tid = threadIdx.x;
    #pragma unroll
    for (int i = 0; i < 16; ++i) {
        int e = tid + i * 256;           // e = k*64 + n, coalesced read
        int k = e >> 6, n = e & 63;
        sW[n * SW_STRIDE + k] = W[e];
    }
    if (tid < 128) sStats[tid] = 0.f;
    __syncthreads();

    int wave = tid >> 5;
    int lane = tid & 31;
    int half = lane >> 4;                // K split: low lanes k0,k0+1; high lanes k0+2,k0+3
    int idx  = lane & 15;                // M row (A) / N col (B,C)
    int tile = blockIdx.x * 8 + wave;

    if (tile < MTILES) {
        const float* zrow = Zin + (tile * 16 + idx) * 64;
        // A fragments for full K=64 (16 steps of K=4), optional fused BN+ReLU
        v2f a[16];
        #pragma unroll
        for (int t = 0; t < 16; ++t) {
            int k = 4 * t + 2 * half;
            v2f v = *(const v2f*)(zrow + k);
            if (doBN) {
                v2f sc = *(const v2f*)(scale + k);
                v2f sh = *(const v2f*)(shift + k);
                v.x = fmaxf(v.x * sc.x + sh.x, 0.f);
                v.y = fmaxf(v.y * sc.y + sh.y, 0.f);
            }
            a[t] = v;
        }

        v8f acc[4];
        #pragma unroll
        for (int nt = 0; nt < 4; ++nt)
            acc[nt] = (v8f){0.f,0.f,0.f,0.f,0.f,0.f,0.f,0.f};

        #pragma unroll
        for (int t = 0; t < 16; ++t) {
            #pragma unroll
            for (int nt = 0; nt < 4; ++nt) {
                int n = nt * 16 + idx;
                v2f b = *(const v2f*)(&sW[n * SW_STRIDE + 4 * t + 2 * half]);
                acc[nt] = __builtin_amdgcn_wmma_f32_16x16x4_f32(
                    false, a[t], false, b, (short)0, acc[nt], false, false);
            }
        }

        #pragma unroll
        for (int nt = 0; nt < 4; ++nt) {
            int n = nt * 16 + idx;
            float bn = bias[n];
            float s = 0.f, sq = 0.f;
            #pragma unroll
            for (int g = 0; g < 8; ++g) {  // C VGPR g -> row M = g + 8*half
                float v = acc[nt][g] + bn;
                Out[(tile * 16 + g + 8 * half) * 64 + n] = v;
                s += v; sq += v * v;
            }
            if (doStats) {
                atomicAdd(&sStats[n], s);        // ds_add_f32
                atomicAdd(&sStats[64 + n], sq);
            }
        }
    }

    if (doStats) {
        __syncthreads();
        if (tid < 128) gatomicAdd(&stats[tid], sStats[tid]);
    }
}

// stats -> (scale, shift); also re-zeros stats for next layer
__global__ void bnprep_kernel(float* __restrict__ stats,
                              const float* __restrict__ gamma, const float* __restrict__ beta,
                              float* __restrict__ scale, float* __restrict__ shift) {
    int t = threadIdx.x;
    if (t < 64) {
        float invN = 1.f / (float)NNODES;
        float mean = stats[t] * invN;
        float var  = stats[64 + t] * invN - mean * mean;
        float sc   = gamma[t] * rsqrtf(var + BN_EPS);
        scale[t] = sc;
        shift[t] = beta[t] - mean * sc;
        stats[t] = 0.f; stats[64 + t] = 0.f;
    }
}

// segment sum over sorted graph ids
__global__ void pool_kernel(const float* __restrict__ h, const long long* __restrict__ batch,
                            float* __restrict__ pool, float* __restrict__ counts) {
    int t  = blockIdx.x * 256 + threadIdx.x;
    int i  = t >> 5;
    int fp = t & 31;
    if (i >= NNODES) return;
    long long g = batch[i];
    float2 v = ((const float2*)(h + (long long)i * 64))[fp];
    float* pp = pool + g * 64 + fp * 2;
    gatomicAdd(pp,     v.x);
    gatomicAdd(pp + 1, v.y);
    if (fp == 0) gatomicAdd(&counts[g], 1.f);
}

__global__ void final_kernel(const float* __restrict__ pool, const float* __restrict__ counts,
                             float* __restrict__ out) {
    int t = blockIdx.x * 256 + threadIdx.x;
    if (t < NGRAPH * 64) out[t] = pool[t] / fmaxf(counts[t >> 6], 1.f);
}

// ---------------- orchestration ----------------

extern "C" void kernel_launch(void* const* d_in, const int* in_sizes, int n_in,
                              void* d_out, int out_size, void* d_ws, size_t ws_size,
                              hipStream_t stream) {
    const float*      x     = (const float*)d_in[0];
    const long long*  ei    = (const long long*)d_in[1];
    const long long*  batch = (const long long*)d_in[2];
    const float *W1_0 = (const float*)d_in[3],  *b1_0 = (const float*)d_in[4];
    const float *g_0  = (const float*)d_in[5],  *be_0 = (const float*)d_in[6];
    const float *W2_0 = (const float*)d_in[7],  *b2_0 = (const float*)d_in[8];
    const float *W1_1 = (const float*)d_in[9],  *b1_1 = (const float*)d_in[10];
    const float *g_1  = (const float*)d_in[11], *be_1 = (const float*)d_in[12];
    const float *W2_1 = (const float*)d_in[13], *b2_1 = (const float*)d_in[14];
    float* out = (float*)d_out;

    const size_t NF = (size_t)NNODES * 64;
    float* z      = (float*)d_ws;      // N x 64  (GIN aggregation input)
    float* h      = z + NF;            // N x 64  (hidden, pre-BN)
    float* y1     = h + NF;            // N x 64  (layer-1 output)
    float* stats  = y1 + NF;           // 128
    float* scale  = stats + 128;       // 64
    float* shift  = scale + 64;        // 64
    float* pool   = shift + 64;        // 64*64
    float* counts = pool + 4096;       // 64

    dim3 B(256);
    int copyG = (int)((NF / 4 + 255) / 256);
    int scatG = (int)(((size_t)NEDGES * 32 + 255) / 256);
    int gemmG = (MTILES + 7) / 8;
    int poolG = (int)(((size_t)NNODES * 32 + 255) / 256);

    zero_kernel<<<16, B, 0, stream>>>(stats, pool, counts);

    // ---- layer 1 ----
    copy_kernel<<<copyG, B, 0, stream>>>((const float4*)x, (float4*)z, (int)(NF / 4));
    scatter_kernel<<<scatG, B, 0, stream>>>(x, ei, z);
    gemm_kernel<<<gemmG, B, 0, stream>>>(z, W1_0, b1_0, nullptr, nullptr, h, stats, 0, 1);
    bnprep_kernel<<<1, 64, 0, stream>>>(stats, g_0, be_0, scale, shift);
    gemm_kernel<<<gemmG, B, 0, stream>>>(h, W2_0, b2_0, scale, shift, y1, stats, 1, 0);

    // ---- layer 2 ----
    copy_kernel<<<copyG, B, 0, stream>>>((const float4*)y1, (float4*)z, (int)(NF / 4));
    scatter_kernel<<<scatG, B, 0, stream>>>(y1, ei, z);
    gemm_kernel<<<gemmG, B, 0, stream>>>(z, W1_1, b1_1, nullptr, nullptr, h, stats, 0, 1);
    bnprep_kernel<<<1, 64, 0, stream>>>(stats, g_1, be_1, scale, shift);
    gemm_kernel<<<gemmG, B, 0, stream>>>(h, W2_1, b2_1, scale, shift, out, stats, 1, 0);

    // ---- global mean pool ----
    pool_kernel<<<poolG, B, 0, stream>>>(out, batch, pool, counts);
    final_kernel<<<16, B, 0, stream>>>(pool, counts, out + NF);
}